// DKWinners_87454124081886
// MI455X (gfx1250) — compile-verified
//
#include <hip/hip_runtime.h>
#include <hip/hip_bf16.h>

// DKWinners: per-group-of-4 maxout mask * x.
// x: [4096, 4096*4] f32 contiguous; group = 4 consecutive floats.
// Streaming: 1x b128 NT load + 1x b128 NT store per group; 4 groups/thread.
// Hot kernel has ZERO range checks (host verifies exact divisibility);
// general kernel uses a block-uniform scalar branch for the tail.

typedef __attribute__((ext_vector_type(4))) float v4f;

#define BLOCK 256u
#define GPT   4u   // groups per thread

__device__ __forceinline__ v4f kwin_group(v4f v) {
    // argmax with first-occurrence tie-break (strict >), then one-hot gate
    float m = v.x;
    int idx = 0;
    if (v.y > m) { m = v.y; idx = 1; }
    if (v.z > m) { m = v.z; idx = 2; }
    if (v.w > m) { m = v.w; idx = 3; }
    v4f r;
    r.x = (idx == 0) ? v.x : 0.0f;
    r.y = (idx == 1) ? v.y : 0.0f;
    r.z = (idx == 2) ? v.z : 0.0f;
    r.w = (idx == 3) ? v.w : 0.0f;
    return r;
}

__device__ __forceinline__ void kwin_do4(const v4f* __restrict__ x,
                                         v4f* __restrict__ out,
                                         unsigned int g0, unsigned int s) {
    const unsigned int g1 = g0 + s;
    const unsigned int g2 = g0 + 2u * s;
    const unsigned int g3 = g0 + 3u * s;
    v4f v0 = __builtin_nontemporal_load(&x[g0]);
    v4f v1 = __builtin_nontemporal_load(&x[g1]);
    v4f v2 = __builtin_nontemporal_load(&x[g2]);
    v4f v3 = __builtin_nontemporal_load(&x[g3]);
    v4f r0 = kwin_group(v0);
    v4f r1 = kwin_group(v1);
    v4f r2 = kwin_group(v2);
    v4f r3 = kwin_group(v3);
    __builtin_nontemporal_store(r0, &out[g0]);
    __builtin_nontemporal_store(r1, &out[g1]);
    __builtin_nontemporal_store(r2, &out[g2]);
    __builtin_nontemporal_store(r3, &out[g3]);
}

// Hot kernel: ngroups is an exact multiple of BLOCK*GPT (true for 4096x16384).
// No compares, no branches: address math -> 4 NT b128 loads -> 4 NT stores.
__global__ void __launch_bounds__(BLOCK)
DKWinners_exact_kernel(const v4f* __restrict__ x, v4f* __restrict__ out) {
    const unsigned int tid     = blockIdx.x * blockDim.x + threadIdx.x;
    const unsigned int nthread = gridDim.x * blockDim.x;
    kwin_do4(x, out, tid, nthread);
}

// General kernel: block-uniform scalar branch picks fast vs tail path.
__global__ void __launch_bounds__(BLOCK)
DKWinners_general_kernel(const v4f* __restrict__ x, v4f* __restrict__ out,
                         unsigned int ngroups) {
    const unsigned int nthread = gridDim.x * blockDim.x;
    // Largest group index touched by ANY thread of this block — scalar-only
    // operands (blockIdx/blockDim/gridDim), so this branch is wave-uniform.
    const unsigned int block_last =
        (blockIdx.x * blockDim.x + (blockDim.x - 1u)) + (GPT - 1u) * nthread;
    const unsigned int tid = blockIdx.x * blockDim.x + threadIdx.x;

    if (block_last < ngroups) {
        kwin_do4(x, out, tid, nthread);
    } else {
        #pragma unroll
        for (unsigned int k = 0; k < GPT; ++k) {
            const unsigned int g = tid + k * nthread;
            if (g < ngroups) {
                v4f v = __builtin_nontemporal_load(&x[g]);
                v4f r = kwin_group(v);
                __builtin_nontemporal_store(r, &out[g]);
            }
        }
    }
}

extern "C" void kernel_launch(void* const* d_in, const int* in_sizes, int n_in,
                              void* d_out, int out_size, void* d_ws, size_t ws_size,
                              hipStream_t stream) {
    (void)n_in; (void)d_ws; (void)ws_size; (void)in_sizes;
    const v4f* x = (const v4f*)d_in[0];
    v4f* out = (v4f*)d_out;

    // out_size = total float elements (same as input); groups of 4.
    const unsigned int ngroups = (unsigned int)(out_size / 4);
    const unsigned int groups_per_block = BLOCK * GPT;  // 1024

    if (ngroups % groups_per_block == 0u) {
        const unsigned int grid = ngroups / groups_per_block;  // 16384 for ref shape
        DKWinners_exact_kernel<<<grid, BLOCK, 0, stream>>>(x, out);
    } else {
        const unsigned int grid = (ngroups + groups_per_block - 1u) / groups_per_block;
        DKWinners_general_kernel<<<grid, BLOCK, 0, stream>>>(x, out, ngroups);
    }
}